// KAN_GAL_52269751992743
// MI455X (gfx1250) — compile-verified
//
#include <hip/hip_runtime.h>
#include <hip/hip_bf16.h>

// Sizes from the reference
#define DIM     64
#define NSEQ    256
#define BATCH   4
#define NB      8
#define KTOT    576    // 512 spline-basis rows + 64 silu/base rows
#define KSTEPS  144    // KTOT / 4 (K per V_WMMA_F32_16X16X4_F32)
#define ASTRIDE 580    // LDS row stride (dwords): 16B aligned, conflict-free frag reads
#define ODIM    128

typedef float v2f __attribute__((ext_vector_type(2)));
typedef float v8f __attribute__((ext_vector_type(8)));

// ---------- fast transcendentals (CDNA5 hardware TRANS ops) ----------
#if __has_builtin(__builtin_amdgcn_tanhf)
__device__ __forceinline__ float fast_tanh(float x) { return __builtin_amdgcn_tanhf(x); }
#else
__device__ __forceinline__ float fast_tanh(float x) { return tanhf(x); }
#endif
#define LOG2E 1.4426950408889634f
// sigmoid: safe branch-free path (used with bounded args)
__device__ __forceinline__ float fast_sigmoid(float x) {
  return __builtin_amdgcn_rcpf(1.f + __builtin_amdgcn_exp2f(-LOG2E * x));
}
__device__ __forceinline__ float fast_exp(float x) {  // x <= 0 in our uses
  return __builtin_amdgcn_exp2f(LOG2E * x);
}

// ---------- wave32 reductions ----------
__device__ __forceinline__ float wave_sum32(float v) {
  v += __shfl_xor(v, 16, 32);
  v += __shfl_xor(v, 8, 32);
  v += __shfl_xor(v, 4, 32);
  v += __shfl_xor(v, 2, 32);
  v += __shfl_xor(v, 1, 32);
  return v;
}
__device__ __forceinline__ float half_sum16(float v) {  // reduce within each 16-lane half
  v += __shfl_xor(v, 8, 32);
  v += __shfl_xor(v, 4, 32);
  v += __shfl_xor(v, 2, 32);
  v += __shfl_xor(v, 1, 32);
  return v;
}

// ---------- Cox-de Boor order-3 B-spline, uniform knots (spacing 0.4) ----------
__device__ __forceinline__ void bspline8(float x, float* outb) {
  const float kn[12] = {-2.2f, -1.8f, -1.4f, -1.0f, -0.6f, -0.2f,
                         0.2f,  0.6f,  1.0f,  1.4f,  1.8f,  2.2f};
  float bb[11];
#pragma unroll
  for (int j = 0; j < 11; j++) bb[j] = (x >= kn[j] && x < kn[j + 1]) ? 1.f : 0.f;
#pragma unroll
  for (int k = 1; k <= 3; k++) {
    const float inv = 2.5f / (float)k;  // 1/(0.4*k)
#pragma unroll
    for (int j = 0; j + k < 11; j++) {
      bb[j] = (x - kn[j]) * inv * bb[j] + (kn[j + k + 1] - x) * inv * bb[j + 1];
    }
  }
#pragma unroll
  for (int j = 0; j < 8; j++) outb[j] = bb[j];
}

// ---------- multi-tile GEMM: D(16 x 16*NT) = A(16xKTOT) * CB2(KTOT x outdim) ----------
// A in LDS, row-major stride ASTRIDE. CB2 layout: elem(k,o) at ((k>>1)*outdim + o)*2 + (k&1).
// A-frag (16x4 f32): lanes 0-15 -> row=l16, K=4s,4s+1 ; lanes 16-31 -> row=l16, K=4s+2,4s+3.
// B-frag (4x16 f32): lanes 0-15 -> col=o,  K=4s,4s+1 ; lanes 16-31 -> col=o,  K=4s+2,4s+3.
// One A-fragment load feeds NT consecutive WMMAs (tiles o = t*16 + l16).
template <int NT>
__device__ __forceinline__ void kan_gemm_multi(const float* __restrict__ A,
                                               const float* __restrict__ cb2,
                                               int outdim, int l16, int half, v8f* acc) {
  const float* arow = A + l16 * ASTRIDE + 2 * half;
  const float* bbase = cb2 + (size_t)(half * outdim + l16) * 2;
#pragma unroll 2
  for (int s = 0; s < KSTEPS; s++) {
    if ((s & 15) == 0)  // stream-ahead hint for the L2-resident B matrix
      __builtin_prefetch(bbase + (size_t)(2 * (s + 16)) * outdim * 2, 0, 1);
    v2f a = *(const v2f*)(arow + 4 * s);
    const float* bp = bbase + (size_t)(2 * s) * outdim * 2;
#pragma unroll
    for (int t = 0; t < NT; t++) {
      v2f b = *(const v2f*)(bp + t * 32);  // next 16 output cols -> +16*2 floats
      acc[t] = __builtin_amdgcn_wmma_f32_16x16x4_f32(false, a, false, b, (short)0,
                                                     acc[t], false, false);
    }
  }
}

// ---------- stage 16 rows of KAN activations (tanh -> bases + silu) into LDS ----------
__device__ __forceinline__ void stage16(float* A, const float* __restrict__ src, int lane) {
  const int i1 = lane, i2 = lane + 32;
  for (int r = 0; r < 16; r++) {
    float x1 = fast_tanh(src[r * DIM + i1]);
    float x2 = fast_tanh(src[r * DIM + i2]);
    A[r * ASTRIDE + 512 + i1] = x1 * fast_sigmoid(x1);
    A[r * ASTRIDE + 512 + i2] = x2 * fast_sigmoid(x2);
    float bs[8];
    bspline8(x1, bs);
    float4 lo = {bs[0], bs[1], bs[2], bs[3]}, hi = {bs[4], bs[5], bs[6], bs[7]};
    *(float4*)(A + r * ASTRIDE + i1 * 8) = lo;
    *(float4*)(A + r * ASTRIDE + i1 * 8 + 4) = hi;
    bspline8(x2, bs);
    float4 lo2 = {bs[0], bs[1], bs[2], bs[3]}, hi2 = {bs[4], bs[5], bs[6], bs[7]};
    *(float4*)(A + r * ASTRIDE + i2 * 8) = lo2;
    *(float4*)(A + r * ASTRIDE + i2 * 8 + 4) = hi2;
  }
}

// ---------- prep: LayerNorm(h) and L2-normalized hn ----------
__global__ __launch_bounds__(32) void lnnorm_kernel(const float* __restrict__ h,
                                                    const float* __restrict__ g,
                                                    const float* __restrict__ be,
                                                    float* __restrict__ ln_h,
                                                    float* __restrict__ hn) {
  const int row = blockIdx.x, lane = threadIdx.x;
  const float* src = h + (size_t)row * DIM;
  float a1 = src[lane], a2 = src[lane + 32];
  float s = wave_sum32(a1 + a2);
  float sq = wave_sum32(a1 * a1 + a2 * a2);
  float mean = s * (1.f / 64.f);
  float var = sq * (1.f / 64.f) - mean * mean;
  float rs = rsqrtf(var + 1e-5f);
  float l1 = (a1 - mean) * rs * g[lane] + be[lane];
  float l2 = (a2 - mean) * rs * g[lane + 32] + be[lane + 32];
  ln_h[(size_t)row * DIM + lane] = l1;
  ln_h[(size_t)row * DIM + lane + 32] = l2;
  float nrm = sqrtf(wave_sum32(l1 * l1 + l2 * l2));
  float inv = 1.f / fmaxf(nrm, 1e-12f);
  hn[(size_t)row * DIM + lane] = l1 * inv;
  hn[(size_t)row * DIM + lane + 32] = l2 * inv;
}

// ---------- prep: fold coeffs into combined B matrix (spline rows + base_w rows) ----------
__global__ void build_cb_kernel(const float* __restrict__ bw, const float* __restrict__ sw,
                                const float* __restrict__ c, float* __restrict__ cb2,
                                int outdim) {
  int idx = blockIdx.x * blockDim.x + threadIdx.x;
  if (idx >= KTOT * outdim) return;
  int k = idx / outdim, o = idx - k * outdim;
  float v;
  if (k < 512) {
    int i = k >> 3, kb = k & 7;
    float cv = c[((size_t)o * DIM + i) * NB + kb];
    cv = fminf(fmaxf(cv, -5.f), 5.f);
    v = cv * sw[(size_t)o * DIM + i];
  } else {
    v = bw[(size_t)o * DIM + (k - 512)];
  }
  cb2[((size_t)(k >> 1) * outdim + o) * 2 + (k & 1)] = v;
}

__global__ void wmean_kernel(const float* __restrict__ W, float* __restrict__ wm) {
  int d = threadIdx.x;
  if (d < DIM) {
    float s = 0.f;
    for (int j = 0; j < NSEQ; j++) s += W[(size_t)d * NSEQ + j];
    wm[d] = s * (1.f / 256.f);
  }
}

// ---------- main WMMA kernel: fused prod -> LN -> KAN -> tanh -> dot(w_mean) ----------
__global__ __launch_bounds__(32) void score_kernel(const float* __restrict__ hn,
                                                   const float* __restrict__ lnpg,
                                                   const float* __restrict__ lnpb,
                                                   const float* __restrict__ cb2a,
                                                   const float* __restrict__ wmean,
                                                   float* __restrict__ scores) {
  __shared__ float A[16 * ASTRIDE];
  const int lane = threadIdx.x;
  const int l16 = lane & 15, half = lane >> 4;
  const int m0 = blockIdx.x * 16, n = blockIdx.y, b = blockIdx.z;
  const int i1 = lane, i2 = lane + 32;
  const float* hnn = hn + (size_t)(b * NSEQ + n) * DIM;
  const float hn1 = hnn[i1], hn2 = hnn[i2];
  const float g1 = lnpg[i1], g2 = lnpg[i2];
  const float c1 = lnpb[i1], c2 = lnpb[i2];

  for (int r = 0; r < 16; r++) {
    const float* hm = hn + (size_t)(b * NSEQ + m0 + r) * DIM;
    float p1 = hn1 * hm[i1], p2 = hn2 * hm[i2];
    float s = wave_sum32(p1 + p2);
    float sq = wave_sum32(p1 * p1 + p2 * p2);
    float mean = s * (1.f / 64.f);
    float var = sq * (1.f / 64.f) - mean * mean;
    float rs = rsqrtf(var + 1e-5f);
    float x1 = fast_tanh((p1 - mean) * rs * g1 + c1);
    float x2 = fast_tanh((p2 - mean) * rs * g2 + c2);
    A[r * ASTRIDE + 512 + i1] = x1 * fast_sigmoid(x1);
    A[r * ASTRIDE + 512 + i2] = x2 * fast_sigmoid(x2);
    float bs[8];
    bspline8(x1, bs);
    float4 lo = {bs[0], bs[1], bs[2], bs[3]}, hi = {bs[4], bs[5], bs[6], bs[7]};
    *(float4*)(A + r * ASTRIDE + i1 * 8) = lo;
    *(float4*)(A + r * ASTRIDE + i1 * 8 + 4) = hi;
    bspline8(x2, bs);
    float4 lo2 = {bs[0], bs[1], bs[2], bs[3]}, hi2 = {bs[4], bs[5], bs[6], bs[7]};
    *(float4*)(A + r * ASTRIDE + i2 * 8) = lo2;
    *(float4*)(A + r * ASTRIDE + i2 * 8 + 4) = hi2;
  }
  __syncthreads();

  v8f acc[4];
#pragma unroll
  for (int t = 0; t < 4; t++) acc[t] = (v8f){0.f, 0.f, 0.f, 0.f, 0.f, 0.f, 0.f, 0.f};
  kan_gemm_multi<4>(A, cb2a, DIM, l16, half, acc);

  float partial[8] = {0.f, 0.f, 0.f, 0.f, 0.f, 0.f, 0.f, 0.f};
#pragma unroll
  for (int t = 0; t < 4; t++) {
    float wm = wmean[t * 16 + l16];
#pragma unroll
    for (int v = 0; v < 8; v++) partial[v] += fast_tanh(acc[t][v]) * wm;
  }
#pragma unroll
  for (int v = 0; v < 8; v++) partial[v] = half_sum16(partial[v]);
  if (l16 == 0) {
    float* so = scores + (size_t)(b * NSEQ + n) * NSEQ + m0 + half * 8;
#pragma unroll
    for (int v = 0; v < 8; v++) so[v] = partial[v];
  }
}

// ---------- softmax + h_att = attn @ ln_h ----------
__global__ __launch_bounds__(256) void attn_kernel(const float* __restrict__ scores,
                                                   const float* __restrict__ ln_h,
                                                   float* __restrict__ h_att) {
  __shared__ float e_l[NSEQ];
  __shared__ float red[8];
  const int bn = blockIdx.x;
  const int b = bn >> 8;
  const int t = threadIdx.x;
  float s = scores[(size_t)bn * NSEQ + t] * 0.125f;  // / (sqrt(64)*TEMP)
  float m = s;
  m = fmaxf(m, __shfl_xor(m, 16, 32));
  m = fmaxf(m, __shfl_xor(m, 8, 32));
  m = fmaxf(m, __shfl_xor(m, 4, 32));
  m = fmaxf(m, __shfl_xor(m, 2, 32));
  m = fmaxf(m, __shfl_xor(m, 1, 32));
  if ((t & 31) == 0) red[t >> 5] = m;
  __syncthreads();
  float M = red[0];
#pragma unroll
  for (int w = 1; w < 8; w++) M = fmaxf(M, red[w]);
  __syncthreads();
  float e = fast_exp(s - M);  // s - M <= 0
  e_l[t] = e;
  float ps = wave_sum32(e);
  if ((t & 31) == 0) red[t >> 5] = ps;
  __syncthreads();
  float S = 0.f;
#pragma unroll
  for (int w = 0; w < 8; w++) S += red[w];
  if (t < DIM) {
    float inv = 1.f / S;
    float acc = 0.f;
    for (int mm = 0; mm < NSEQ; mm++)
      acc += e_l[mm] * ln_h[(size_t)(b * NSEQ + mm) * DIM + t];
    h_att[(size_t)bn * DIM + t] = acc * inv;
  }
}

// ---------- output KAN layers (kp on h_att, kd on ln_h) + BatchNorm, via WMMA ----------
__global__ __launch_bounds__(32) void outkan_kernel(
    const float* __restrict__ h_att, const float* __restrict__ ln_h,
    const float* __restrict__ cb2p, const float* __restrict__ cb2d,
    const float* __restrict__ bng, const float* __restrict__ bnb,
    const float* __restrict__ bnm, const float* __restrict__ bnv,
    float* __restrict__ outp) {
  __shared__ float A[16 * ASTRIDE];
  const int lane = threadIdx.x;
  const int l16 = lane & 15, half = lane >> 4;
  const int row0 = blockIdx.x * 16;

  stage16(A, h_att + (size_t)row0 * DIM, lane);
  __syncthreads();
  v8f acc2[8];
#pragma unroll
  for (int t = 0; t < 8; t++) acc2[t] = (v8f){0.f, 0.f, 0.f, 0.f, 0.f, 0.f, 0.f, 0.f};
  kan_gemm_multi<8>(A, cb2p, ODIM, l16, half, acc2);
  __syncthreads();
  stage16(A, ln_h + (size_t)row0 * DIM, lane);
  __syncthreads();
  v8f acc3[8];
#pragma unroll
  for (int t = 0; t < 8; t++) acc3[t] = (v8f){0.f, 0.f, 0.f, 0.f, 0.f, 0.f, 0.f, 0.f};
  kan_gemm_multi<8>(A, cb2d, ODIM, l16, half, acc3);
#pragma unroll
  for (int t = 0; t < 8; t++) {
    int o = t * 16 + l16;
    float scale = bng[o] * rsqrtf(bnv[o] + 1e-5f);
    float mu = bnm[o], bo = bnb[o];
#pragma unroll
    for (int v = 0; v < 8; v++) {
      float val = (acc2[t][v] + acc3[t][v] - mu) * scale + bo;
      outp[(size_t)(row0 + half * 8 + v) * ODIM + o] = val;
    }
  }
}

extern "C" void kernel_launch(void* const* d_in, const int* in_sizes, int n_in,
                              void* d_out, int out_size, void* d_ws, size_t ws_size,
                              hipStream_t stream) {
  (void)in_sizes; (void)n_in; (void)out_size; (void)ws_size;
  const float* h       = (const float*)d_in[0];
  const float* ln_in_g = (const float*)d_in[1];
  const float* ln_in_b = (const float*)d_in[2];
  const float* ln_p_g  = (const float*)d_in[3];
  const float* ln_p_b  = (const float*)d_in[4];
  const float* ka_bw   = (const float*)d_in[5];
  const float* ka_sw   = (const float*)d_in[6];
  const float* ka_c    = (const float*)d_in[7];
  const float* W_att   = (const float*)d_in[8];
  const float* kp_bw   = (const float*)d_in[9];
  const float* kp_sw   = (const float*)d_in[10];
  const float* kp_c    = (const float*)d_in[11];
  const float* kd_bw   = (const float*)d_in[12];
  const float* kd_sw   = (const float*)d_in[13];
  const float* kd_c    = (const float*)d_in[14];
  const float* bn_g    = (const float*)d_in[15];
  const float* bn_b    = (const float*)d_in[16];
  const float* bn_mean = (const float*)d_in[17];
  const float* bn_var  = (const float*)d_in[18];
  float* outp = (float*)d_out;

  // Workspace layout (floats): ~2.6 MB total
  float* ws     = (float*)d_ws;
  float* ln_h   = ws;                 // 4*256*64   = 65536
  float* hnv    = ln_h + 65536;       // 65536
  float* h_att  = hnv + 65536;        // 65536
  float* wmean  = h_att + 65536;      // 64
  float* cb2a   = wmean + 64;         // 576*64     = 36864
  float* cb2p   = cb2a + 36864;       // 576*128    = 73728
  float* cb2d   = cb2p + 73728;       // 73728
  float* scores = cb2d + 73728;       // 4*256*256  = 262144

  lnnorm_kernel<<<BATCH * NSEQ, 32, 0, stream>>>(h, ln_in_g, ln_in_b, ln_h, hnv);
  build_cb_kernel<<<(KTOT * DIM + 255) / 256, 256, 0, stream>>>(ka_bw, ka_sw, ka_c, cb2a, DIM);
  build_cb_kernel<<<(KTOT * ODIM + 255) / 256, 256, 0, stream>>>(kp_bw, kp_sw, kp_c, cb2p, ODIM);
  build_cb_kernel<<<(KTOT * ODIM + 255) / 256, 256, 0, stream>>>(kd_bw, kd_sw, kd_c, cb2d, ODIM);
  wmean_kernel<<<1, 64, 0, stream>>>(W_att, wmean);

  dim3 sg(NSEQ / 16, NSEQ, BATCH);  // 16384 single-wave workgroups
  score_kernel<<<sg, 32, 0, stream>>>(hnv, ln_p_g, ln_p_b, cb2a, wmean, scores);
  attn_kernel<<<BATCH * NSEQ, 256, 0, stream>>>(scores, ln_h, h_att);
  outkan_kernel<<<(BATCH * NSEQ) / 16, 32, 0, stream>>>(h_att, ln_h, cb2p, cb2d,
                                                        bn_g, bn_b, bn_mean, bn_var, outp);
}